// hnode_prompt_layer_feature_cat_edge_21534966022316
// MI455X (gfx1250) — compile-verified
//
#include <hip/hip_runtime.h>

#ifndef __has_builtin
#define __has_builtin(x) 0
#endif

#define GLOBAL_AS __attribute__((address_space(1)))
#define LDS_AS    __attribute__((address_space(3)))

typedef int v4i __attribute__((vector_size(16)));   // matches clang builtin sig "V4i"

static constexpr int kDFt  = 128;               // node/prompt feature dim
static constexpr int kDE   = 32;                // edge feature dim
static constexpr int kDOut = kDFt + kDE;        // 160

// ---------------------------------------------------------------------------
// Zero the output buffer (atomics accumulate into it; harness poisons d_out).
// ---------------------------------------------------------------------------
__global__ __launch_bounds__(256) void zero_out_kernel(float* __restrict__ o, int n) {
    const int n4 = n >> 2;
    float4* __restrict__ o4 = reinterpret_cast<float4*>(o);
    int i = blockIdx.x * blockDim.x + threadIdx.x;
    const int st = gridDim.x * blockDim.x;
    const float4 z = make_float4(0.f, 0.f, 0.f, 0.f);
    for (; i < n4; i += st) o4[i] = z;
    // scalar tail (out_size = 8,000,000 is divisible by 4, but stay general)
    if (blockIdx.x == 0 && threadIdx.x == 0) {
        for (int r = n4 << 2; r < n; ++r) o[r] = 0.f;
    }
}

// ---------------------------------------------------------------------------
// Wave-per-edge gather/scale/scatter.
//   lane L of the wave owns features [4L, 4L+4) of the 128-wide half and
//   feature L of the 32-wide edge half.  One b128 load fetches the whole
//   512B source row per wave; 5 non-returning f32 atomics per lane scatter
//   into the destination row (resident in the 192MB L2).
// ---------------------------------------------------------------------------
__global__ __launch_bounds__(256) void edge_scatter_kernel(
    const float* __restrict__ ge,      // [N, 128]
    const float* __restrict__ efeat,   // [E, 32]
    const float* __restrict__ w,       // [1, 128]
    const int*   __restrict__ src,     // [E]
    const int*   __restrict__ dst,     // [E]
    float*       __restrict__ out,     // [N, 160]
    int nEdges)
{
    __shared__ float wlds[kDFt];
    const int lane = threadIdx.x & 31;

#if __has_builtin(__builtin_amdgcn_global_load_async_to_lds_b128)
    // Stage the 512B prompt vector with one async copy per lane:
    // 32 lanes x 16B = 128 floats in a single GLOBAL_LOAD_ASYNC_TO_LDS_B128.
    if (threadIdx.x < 32) {
        __builtin_amdgcn_global_load_async_to_lds_b128(
            (GLOBAL_AS v4i*)(w + lane * 4),
            (LDS_AS   v4i*)(&wlds[lane * 4]),
            0, 0);
    }
#if __has_builtin(__builtin_amdgcn_s_wait_asynccnt)
    __builtin_amdgcn_s_wait_asynccnt(0);
#else
    asm volatile("s_wait_asynccnt 0x0" ::: "memory");
#endif
#else
    if (threadIdx.x < kDFt) wlds[threadIdx.x] = w[threadIdx.x];
#endif
    __syncthreads();

    const float4 wl = *reinterpret_cast<const float4*>(&wlds[lane * 4]);

    const int wavesPerBlk = blockDim.x >> 5;
    const int waveId = blockIdx.x * wavesPerBlk + (threadIdx.x >> 5);
    const int nWaves = gridDim.x * wavesPerBlk;

    for (int e = waveId; e < nEdges; e += nWaves) {
        const int en = e + nWaves;
        if (en < nEdges) {
            // global_prefetch_b8 on the streaming index / edge-feature arrays
            __builtin_prefetch(src + en, 0, 0);
            __builtin_prefetch(dst + en, 0, 0);
            __builtin_prefetch(efeat + (size_t)en * kDE + lane, 0, 0);
        }

        const int s = __builtin_amdgcn_readfirstlane(src[e]);
        const int d = __builtin_amdgcn_readfirstlane(dst[e]);

        const float4 sf =
            *reinterpret_cast<const float4*>(ge + (size_t)s * kDFt + lane * 4);
        const float ev = efeat[(size_t)e * kDE + lane];

        float4 m;
        m.x = sf.x * wl.x;
        m.y = sf.y * wl.y;
        m.z = sf.z * wl.z;
        m.w = sf.w * wl.w;

        float* ob = out + (size_t)d * kDOut;
        // non-returning hardware fp32 atomics (global_atomic_add_f32)
        unsafeAtomicAdd(ob + lane * 4 + 0, m.x);
        unsafeAtomicAdd(ob + lane * 4 + 1, m.y);
        unsafeAtomicAdd(ob + lane * 4 + 2, m.z);
        unsafeAtomicAdd(ob + lane * 4 + 3, m.w);
        unsafeAtomicAdd(ob + kDFt + lane, ev);
    }
}

// ---------------------------------------------------------------------------
extern "C" void kernel_launch(void* const* d_in, const int* in_sizes, int n_in,
                              void* d_out, int out_size, void* d_ws, size_t ws_size,
                              hipStream_t stream) {
    (void)n_in; (void)d_ws; (void)ws_size;

    const float* ge    = (const float*)d_in[0];  // graph_embedding [N,128]
    const float* efeat = (const float*)d_in[1];  // e_feat          [E,32]
    const float* w     = (const float*)d_in[2];  // weight          [1,128]
    const int*   src   = (const int*)  d_in[3];  // src_idx         [E]
    const int*   dst   = (const int*)  d_in[4];  // dst_idx         [E]
    float*       out   = (float*)d_out;          // [N,160]

    const int nEdges = in_sizes[3];

    zero_out_kernel<<<2048, 256, 0, stream>>>(out, out_size);
    edge_scatter_kernel<<<1536, 256, 0, stream>>>(ge, efeat, w, src, dst, out, nEdges);
}